// ETracking_Net_86526411145630
// MI455X (gfx1250) — compile-verified
//
#include <hip/hip_runtime.h>
#include <hip/hip_bf16.h>
#include <math.h>

#define BB   2
#define PN   16384
#define SS   4096
#define KN   8
#define NNE  (SS*KN)   // 32768 (s,k) pairs per batch

typedef __attribute__((ext_vector_type(2))) float v2f;
typedef __attribute__((ext_vector_type(8))) float v8f;

__device__ __forceinline__ v8f wmma4(v2f a, v2f b, v8f c) {
  // V_WMMA_F32_16X16X4_F32 : D(16x16) = A(16x4) x B(4x16) + C
  return __builtin_amdgcn_wmma_f32_16x16x4_f32(false, a, false, b, (short)0, c, false, false);
}

// ---------------- K0: per-batch min/max of x[:, 0:2, :] ----------------
__global__ void k_minmax(const float* __restrict__ x, float* __restrict__ mm) {
  int b = blockIdx.x >> 1, d = blockIdx.x & 1;
  const float* p = x + (size_t)(b*5 + d)*PN;
  float mn = 3.0e38f, mx = -3.0e38f;
  for (int i = threadIdx.x; i < PN; i += blockDim.x) {
    float v = p[i]; mn = fminf(mn, v); mx = fmaxf(mx, v);
  }
  __shared__ float smn[256], smx[256];
  smn[threadIdx.x] = mn; smx[threadIdx.x] = mx;
  __syncthreads();
  for (int s = 128; s > 0; s >>= 1) {
    if ((int)threadIdx.x < s) {
      smn[threadIdx.x] = fminf(smn[threadIdx.x], smn[threadIdx.x + s]);
      smx[threadIdx.x] = fmaxf(smx[threadIdx.x], smx[threadIdx.x + s]);
    }
    __syncthreads();
  }
  if (threadIdx.x == 0) { mm[b*4 + d*2] = smn[0]; mm[b*4 + d*2 + 1] = smx[0]; }
}

// ---------------- K1: KNN(k=1) of grid points vs x[:, :2, :] ----------------
__global__ void k_knn1(const float* __restrict__ x, const float* __restrict__ mm,
                       int* __restrict__ idx1) {
  int wave = (blockIdx.x * blockDim.x + threadIdx.x) >> 5;
  int lane = threadIdx.x & 31;
  int b = wave / SS, s = wave % SS;
  float mn0 = mm[b*4+0], mx0 = mm[b*4+1], mn1 = mm[b*4+2], mx1 = mm[b*4+3];
  float psx = (float)(s & 63) * (1.0f/64.0f) * (mx0 - mn0) + mn0;
  float psy = (float)(s >> 6) * (1.0f/64.0f) * (mx1 - mn1) + mn1;
  const float* x0 = x + (size_t)(b*5+0)*PN;
  const float* x1c = x + (size_t)(b*5+1)*PN;
  float bs = -3.0e38f; int bi = 0x7fffffff;
  for (int p = lane; p < PN; p += 32) {
    float dx = x0[p] - psx, dy = x1c[p] - psy;
    float sc = -(dx*dx + dy*dy);
    if (sc > bs) { bs = sc; bi = p; }   // increasing p: strict '>' keeps lowest index on ties
  }
  for (int off = 16; off >= 1; off >>= 1) {
    float os = __shfl_xor(bs, off);
    int   oi = __shfl_xor(bi, off);
    if (os > bs || (os == bs && oi < bi)) { bs = os; bi = oi; }
  }
  if (lane == 0) idx1[b*SS + s] = bi;
}

// ---------------- K2a: pack (x,y,z,|p|^2) per point as float4 ----------------
__global__ void k_pack(const float* __restrict__ x, float4* __restrict__ pt) {
  int i = blockIdx.x * blockDim.x + threadIdx.x;   // B*PN
  int b = i / PN, p = i % PN;
  float a0 = x[(size_t)(b*5+0)*PN+p], a1 = x[(size_t)(b*5+1)*PN+p], a2 = x[(size_t)(b*5+2)*PN+p];
  pt[i] = make_float4(a0, a1, a2, a0*a0 + a1*a1 + a2*a2);
}

// ---------------- K2b: gather sampled 3D coords ----------------
__global__ void k_sel3(const float* __restrict__ x, const int* __restrict__ idx1,
                       float* __restrict__ sel3) {
  int i = blockIdx.x * blockDim.x + threadIdx.x;   // B*SS
  int b = i / SS, s = i % SS;
  int p = idx1[i];
  for (int d = 0; d < 3; ++d) sel3[(size_t)(b*3+d)*SS + s] = x[(size_t)(b*5+d)*PN + p];
}

// ---------------- top-8 insertion helper (fully unrolled, stays in VGPRs) ----------------
#define TOP8_INSERT(sc, pi, ts, ti)                                               \
  if ((sc) > ts[7] || ((sc) == ts[7] && (pi) < ti[7])) {                          \
    ts[7] = (sc); ti[7] = (pi);                                                   \
    _Pragma("unroll")                                                             \
    for (int _j = 7; _j >= 1; --_j) {                                             \
      bool _sw = (ts[_j] > ts[_j-1]) || (ts[_j] == ts[_j-1] && ti[_j] < ti[_j-1]);\
      float _tms = _sw ? ts[_j-1] : ts[_j];                                       \
      int   _tmi = _sw ? ti[_j-1] : ti[_j];                                       \
      ts[_j-1] = _sw ? ts[_j] : ts[_j-1];                                         \
      ti[_j-1] = _sw ? ti[_j] : ti[_j-1];                                         \
      ts[_j] = _tms; ti[_j] = _tmi;                                               \
    }                                                                             \
  }

// process one WMMA result: cheap max-tree filter, then (rare) insertion
#define PROC8(dv, baseidx)                                                        \
  { float _mx = fmaxf(fmaxf(fmaxf(dv[0],dv[1]), fmaxf(dv[2],dv[3])),             \
                      fmaxf(fmaxf(dv[4],dv[5]), fmaxf(dv[6],dv[7])));            \
    if (_mx >= ts[7]) {                                                           \
      _Pragma("unroll")                                                           \
      for (int _r = 0; _r < 8; ++_r) {                                            \
        float _sc = dv[_r]; int _pi = (baseidx) + _r;                             \
        TOP8_INSERT(_sc, _pi, ts, ti)                                             \
      }                                                                           \
    }                                                                             \
  }

// ---------------- K3: KNN(k=8) via WMMA distance tiles ----------------
// Score = 2 q.p - |p|^2  (= pd + |q|^2, rank-equivalent per query)
__global__ void __launch_bounds__(128) k_knn2(const float4* __restrict__ pt,
                                              const float* __restrict__ sel3,
                                              int* __restrict__ idx2) {
  int wave = blockIdx.x * 4 + ((int)threadIdx.x >> 5);   // 512 waves total
  int lane = threadIdx.x & 31;
  int lq = lane & 15;
  bool hi = lane >= 16;
  int b = wave >> 8;          // 256 query tiles per batch
  int tile = wave & 255;
  int q = tile * 16 + lq;
  float qx = sel3[(size_t)(b*3+0)*SS + q];
  float qy = sel3[(size_t)(b*3+1)*SS + q];
  float qz = sel3[(size_t)(b*3+2)*SS + q];
  v2f bv;                      // B(4x16): vgpr r, lane l -> K = r + 2*(l>=16), N = l%16
  bv.x = hi ? 2.0f*qz : 2.0f*qx;
  bv.y = hi ? -1.0f   : 2.0f*qy;
  const float4* ptb = pt + (size_t)b*PN;
  float ts[8]; int ti[8];
  #pragma unroll
  for (int j = 0; j < 8; ++j) { ts[j] = -3.0e38f; ti[j] = 0x7fffffff; }
  int ibase = hi ? 8 : 0;
  for (int c0 = 0; c0 < PN; c0 += 64) {
    __builtin_prefetch(ptb + c0 + 256 + lq, 0, 1);   // ~4 iterations ahead
    // 4 x global_load_b128 in flight, then 4 independent WMMAs
    float4 p0 = ptb[c0      + lq];
    float4 p1 = ptb[c0 + 16 + lq];
    float4 p2 = ptb[c0 + 32 + lq];
    float4 p3 = ptb[c0 + 48 + lq];
    v2f a0, a1, a2, a3;        // A(16x4): lane l -> row M=l%16, K={0,1} or {2,3}
    a0.x = hi ? p0.z : p0.x;  a0.y = hi ? p0.w : p0.y;
    a1.x = hi ? p1.z : p1.x;  a1.y = hi ? p1.w : p1.y;
    a2.x = hi ? p2.z : p2.x;  a2.y = hi ? p2.w : p2.y;
    a3.x = hi ? p3.z : p3.x;  a3.y = hi ? p3.w : p3.y;
    v8f z = {};
    v8f d0 = wmma4(a0, bv, z); // D: vgpr r, lane l -> point M = r + 8*(l>=16), query N = l%16
    v8f d1 = wmma4(a1, bv, z);
    v8f d2 = wmma4(a2, bv, z);
    v8f d3 = wmma4(a3, bv, z);
    PROC8(d0, c0      + ibase)
    PROC8(d1, c0 + 16 + ibase)
    PROC8(d2, c0 + 32 + ibase)
    PROC8(d3, c0 + 48 + ibase)
  }
  // merge lane pair (l <-> l+16): snapshot partner first, then insert
  float os_[8]; int oi_[8];
  #pragma unroll
  for (int j = 0; j < 8; ++j) { os_[j] = __shfl_xor(ts[j], 16); oi_[j] = __shfl_xor(ti[j], 16); }
  #pragma unroll
  for (int j = 0; j < 8; ++j) { float sc = os_[j]; int pi = oi_[j]; TOP8_INSERT(sc, pi, ts, ti) }
  if (!hi) {
    #pragma unroll
    for (int j = 0; j < 8; ++j) idx2[(size_t)(b*8 + j)*SS + q] = ti[j];  // [b][k][s] layout
  }
}

// ---------------- K4: scrambled gather (reference reshapes (k,S) flat as (S,k)) ----------------
__global__ void k_gather(const float* __restrict__ x, const int* __restrict__ idx2,
                         float* __restrict__ x1w) {
  int i = blockIdx.x*blockDim.x + threadIdx.x;   // B*NNE
  int b = i / NNE, t = i % NNE;
  int p = idx2[(size_t)b*NNE + t];               // flat (8,S) read linearly == reference bug
  #pragma unroll
  for (int c = 0; c < 5; ++c)
    x1w[(size_t)(b*5+c)*NNE + t] = x[(size_t)(b*5+c)*PN + p];
}

// ---------------- K5w: padded 128x16 weight block ----------------
__global__ void k_wpad(const float* __restrict__ conv_w, const float* __restrict__ pe_w,
                       float* __restrict__ wpad) {
  int i = blockIdx.x*blockDim.x + threadIdx.x;   // 2048
  int r = i >> 4, c = i & 15;
  float v = 0.0f;
  if (r < 64) { if (c < 4) v = conv_w[r*4 + c]; }
  else        { if (c >= 4 && c < 14) v = pe_w[(r-64)*10 + (c-4)]; }
  wpad[i] = v;
}

// ---------------- K5a: build 16-row GEMM input (x1[0:4] ++ PA,PC,diff,temp ++ pad) ----------------
__global__ void k_in16(const float* __restrict__ x1w, float* __restrict__ in16) {
  int i = blockIdx.x*blockDim.x + threadIdx.x;   // B*NNE
  int b = i / NNE, t = i % NNE;
  int t0 = (t >> 3) << 3;                        // k = 0 element of this sample
  float pa0 = x1w[(size_t)(b*5+0)*NNE + t];
  float pa1 = x1w[(size_t)(b*5+1)*NNE + t];
  float pa2 = x1w[(size_t)(b*5+2)*NNE + t];
  float c3  = x1w[(size_t)(b*5+3)*NNE + t];
  float pc0 = x1w[(size_t)(b*5+0)*NNE + t0];
  float pc1 = x1w[(size_t)(b*5+1)*NNE + t0];
  float pc2 = x1w[(size_t)(b*5+2)*NNE + t0];
  float d0 = pa0-pc0, d1 = pa1-pc1, d2 = pa2-pc2;
  float tmp = sqrtf(d0*d0 + d1*d1 + d2*d2);
  float* o = in16 + (size_t)b*16*NNE + t;
  o[0*(size_t)NNE]=pa0;  o[1*(size_t)NNE]=pa1;  o[2*(size_t)NNE]=pa2;  o[3*(size_t)NNE]=c3;
  o[4*(size_t)NNE]=pa0;  o[5*(size_t)NNE]=pa1;  o[6*(size_t)NNE]=pa2;
  o[7*(size_t)NNE]=pc0;  o[8*(size_t)NNE]=pc1;  o[9*(size_t)NNE]=pc2;
  o[10*(size_t)NNE]=d0;  o[11*(size_t)NNE]=d1;  o[12*(size_t)NNE]=d2;
  o[13*(size_t)NNE]=tmp; o[14*(size_t)NNE]=0.0f; o[15*(size_t)NNE]=0.0f;
}

// ---------------- K5b: 128x16 x 16xN conv GEMM via WMMA (biases cancel in BN) ----------------
__global__ void __launch_bounds__(128) k_conv128(const float* __restrict__ wpad,
                                                 const float* __restrict__ in16,
                                                 float* __restrict__ feat) {
  int wave = blockIdx.x*4 + ((int)threadIdx.x >> 5);
  int lane = threadIdx.x & 31; int lq = lane & 15; bool hi = lane >= 16;
  int b = wave / (8*2048); int rem = wave % (8*2048);
  int mt = rem / 2048, nt = rem % 2048;
  int n = nt*16 + lq;
  int koff = hi ? 2 : 0;
  const float* inb = in16 + (size_t)b*16*NNE;
  v8f acc = {};
  #pragma unroll
  for (int kc = 0; kc < 4; ++kc) {
    int k0 = kc*4 + koff;
    v2f av = *(const v2f*)(wpad + (size_t)(mt*16+lq)*16 + k0);   // global_load_b64
    v2f bvv;
    bvv.x = inb[(size_t)k0*NNE + n];
    bvv.y = inb[(size_t)(k0+1)*NNE + n];
    acc = wmma4(av, bvv, acc);
  }
  float* ob = feat + (size_t)b*128*NNE;
  int mbase = mt*16 + (hi ? 8 : 0);
  #pragma unroll
  for (int r = 0; r < 8; ++r) ob[(size_t)(mbase+r)*NNE + n] = acc[r];
}

// ---------------- K6: BN stats per channel ----------------
__global__ void k_stats(const float* __restrict__ feat, float* __restrict__ stats) {
  int c = blockIdx.x;   // 128
  float s1 = 0.0f, s2 = 0.0f;
  for (int i = threadIdx.x; i < BB*NNE; i += blockDim.x) {
    int b = i / NNE, n = i % NNE;
    float v = feat[((size_t)b*128 + c)*NNE + n];
    s1 += v; s2 += v*v;
  }
  __shared__ float a1[256], a2[256];
  a1[threadIdx.x] = s1; a2[threadIdx.x] = s2;
  __syncthreads();
  for (int s = 128; s > 0; s >>= 1) {
    if ((int)threadIdx.x < s) { a1[threadIdx.x] += a1[threadIdx.x+s]; a2[threadIdx.x] += a2[threadIdx.x+s]; }
    __syncthreads();
  }
  if (threadIdx.x == 0) {
    float inv = 1.0f / (float)(BB*NNE);
    float mean = a1[0] * inv;
    stats[c*2]   = mean;
    stats[c*2+1] = a2[0]*inv - mean*mean;
  }
}

// ---------------- K7: BN apply + lrelu(0.2), in place ----------------
__global__ void k_bn(float* __restrict__ feat, const float* __restrict__ stats,
                     const float* __restrict__ bn_g, const float* __restrict__ bn_b,
                     const float* __restrict__ bn2_g, const float* __restrict__ bn2_b) {
  size_t total = (size_t)BB*128*NNE;
  for (size_t i = (size_t)blockIdx.x*blockDim.x + threadIdx.x; i < total;
       i += (size_t)gridDim.x*blockDim.x) {
    int c = (int)((i / NNE) % 128);
    float g  = (c < 64) ? bn_g[c] : bn2_g[c-64];
    float be = (c < 64) ? bn_b[c] : bn2_b[c-64];
    float mean = stats[c*2], var = stats[c*2+1];
    float y = g * (feat[i] - mean) * rsqrtf(var + 1e-5f) + be;
    feat[i] = (y >= 0.0f) ? y : 0.2f*y;
  }
}

// ---------------- K8: attention logits = att2 . lrelu(att1 @ feat + b1) + b2 (WMMA) ------------
__global__ void __launch_bounds__(128) k_att(const float* __restrict__ feat,
                                             const float* __restrict__ att1_w,
                                             const float* __restrict__ att1_b,
                                             const float* __restrict__ att2_w,
                                             const float* __restrict__ att2_b,
                                             float* __restrict__ logits) {
  int wave = blockIdx.x*4 + ((int)threadIdx.x >> 5);   // B*2048 column tiles
  int lane = threadIdx.x & 31; int lq = lane & 15; bool hi = lane >= 16;
  int b = wave / 2048, nt = wave % 2048;
  int n = nt*16 + lq;
  int koff = hi ? 2 : 0;
  const float* fb = feat + (size_t)b*128*NNE;
  float partial = 0.0f;
  #pragma unroll
  for (int mt = 0; mt < 2; ++mt) {
    v8f acc = {};
    #pragma unroll
    for (int kc = 0; kc < 32; ++kc) {
      int k0 = kc*4 + koff;
      v2f av = *(const v2f*)(att1_w + (size_t)(mt*16+lq)*128 + k0);  // b64 load
      v2f bvv;
      bvv.x = fb[(size_t)k0*NNE + n];
      bvv.y = fb[(size_t)(k0+1)*NNE + n];
      acc = wmma4(av, bvv, acc);
    }
    int mbase = mt*16 + (hi ? 8 : 0);
    #pragma unroll
    for (int r = 0; r < 8; ++r) {
      float h = acc[r] + att1_b[mbase + r];
      h = (h >= 0.0f) ? h : 0.2f*h;
      partial += att2_w[mbase + r] * h;
    }
  }
  partial += __shfl_xor(partial, 16);
  if (!hi) logits[(size_t)b*NNE + n] = partial + att2_b[0];
}

// ---------------- K9: softmax over k + weighted channel sum ----------------
__global__ void k_soft(const float* __restrict__ feat, const float* __restrict__ logits,
                       float* __restrict__ fsum) {
  int bs = blockIdx.x; int b = bs / SS, s = bs % SS;
  __shared__ float sl[8];
  if (threadIdx.x < 8) sl[threadIdx.x] = logits[(size_t)b*NNE + s*8 + threadIdx.x];
  __syncthreads();
  float m = sl[0];
  #pragma unroll
  for (int k = 1; k < 8; ++k) m = fmaxf(m, sl[k]);
  float e[8]; float sum = 0.0f;
  #pragma unroll
  for (int k = 0; k < 8; ++k) { e[k] = expf(sl[k]-m); sum += e[k]; }
  float inv = 1.0f/sum;
  int c = threadIdx.x;   // 128 channels
  const float* f = feat + ((size_t)b*128 + c)*NNE + s*8;
  float acc = 0.0f;
  #pragma unroll
  for (int k = 0; k < 8; ++k) acc += f[k]*e[k];
  fsum[((size_t)b*128 + c)*SS + s] = acc*inv;
}

// ---------------- K10: br1@fsum + br2@x1[:, :4, :, 0] + lrelu(0.01) (WMMA) ----------------
__global__ void __launch_bounds__(128) k_final(const float* __restrict__ fsum,
                                               const float* __restrict__ br1_w,
                                               const float* __restrict__ br1_b,
                                               const float* __restrict__ br2_w,
                                               const float* __restrict__ br2_b,
                                               const float* __restrict__ x1w,
                                               float* __restrict__ out) {
  int wave = blockIdx.x*4 + ((int)threadIdx.x >> 5);   // B*8*256 tiles
  int lane = threadIdx.x & 31; int lq = lane & 15; bool hi = lane >= 16;
  int b = wave / (8*256); int rem = wave % (8*256);
  int mt = rem / 256, st = rem % 256;
  int s = st*16 + lq;
  int koff = hi ? 2 : 0;
  const float* fb = fsum + (size_t)b*128*SS;
  v8f acc = {};
  #pragma unroll
  for (int kc = 0; kc < 32; ++kc) {
    int k0 = kc*4 + koff;
    v2f av = *(const v2f*)(br1_w + (size_t)(mt*16+lq)*128 + k0);   // b64 load
    v2f bvv;
    bvv.x = fb[(size_t)k0*SS + s];
    bvv.y = fb[(size_t)(k0+1)*SS + s];
    acc = wmma4(av, bvv, acc);
  }
  float h0 = x1w[(size_t)(b*5+0)*NNE + s*8];
  float h1 = x1w[(size_t)(b*5+1)*NNE + s*8];
  float h2 = x1w[(size_t)(b*5+2)*NNE + s*8];
  float h3 = x1w[(size_t)(b*5+3)*NNE + s*8];
  int mbase = mt*16 + (hi ? 8 : 0);
  #pragma unroll
  for (int r = 0; r < 8; ++r) {
    int m = mbase + r;
    float v = acc[r] + br1_b[m] + br2_b[m]
            + br2_w[m*4+0]*h0 + br2_w[m*4+1]*h1 + br2_w[m*4+2]*h2 + br2_w[m*4+3]*h3;
    v = (v >= 0.0f) ? v : 0.01f*v;
    out[((size_t)b*133 + 5 + m)*SS + s] = v;
  }
}

// ---------------- K11: output head channels 0..4 = x1[:, :5, :, 0] ----------------
__global__ void k_head(const float* __restrict__ x1w, float* __restrict__ out) {
  int i = blockIdx.x*blockDim.x + threadIdx.x;   // B*5*SS
  int b = i / (5*SS); int rem = i % (5*SS); int c = rem / SS, s = rem % SS;
  out[((size_t)b*133 + c)*SS + s] = x1w[(size_t)(b*5+c)*NNE + s*8];
}

extern "C" void kernel_launch(void* const* d_in, const int* in_sizes, int n_in,
                              void* d_out, int out_size, void* d_ws, size_t ws_size,
                              hipStream_t stream) {
  (void)in_sizes; (void)n_in; (void)out_size; (void)ws_size;
  const float* x      = (const float*)d_in[0];
  const float* pe_w   = (const float*)d_in[1];
  const float* bn2_g  = (const float*)d_in[3];
  const float* bn2_b  = (const float*)d_in[4];
  const float* conv_w = (const float*)d_in[5];
  const float* bn_g   = (const float*)d_in[7];
  const float* bn_b   = (const float*)d_in[8];
  const float* att1_w = (const float*)d_in[9];
  const float* att1_b = (const float*)d_in[10];
  const float* att2_w = (const float*)d_in[11];
  const float* att2_b = (const float*)d_in[12];
  const float* br1_w  = (const float*)d_in[13];
  const float* br1_b  = (const float*)d_in[14];
  const float* br2_w  = (const float*)d_in[15];
  const float* br2_b  = (const float*)d_in[16];
  float* out = (float*)d_out;

  float* ws = (float*)d_ws;
  size_t o = 0;
  float* mm    = ws + o; o += 64;
  int*   idx1  = (int*)(ws + o); o += (size_t)BB*SS;
  float4* pt   = (float4*)(ws + o); o += (size_t)BB*PN*4;   // 16B-aligned (offset 33024 B)
  float* sel3  = ws + o; o += (size_t)BB*3*SS;
  int*   idx2  = (int*)(ws + o); o += (size_t)BB*8*SS;
  float* x1w   = ws + o; o += (size_t)BB*5*NNE;
  float* wpad  = ws + o; o += 2048;
  float* in16  = ws + o; o += (size_t)BB*16*NNE;
  float* feat  = ws + o; o += (size_t)BB*128*NNE;
  float* stats = ws + o; o += 256;
  float* logit = ws + o; o += (size_t)BB*NNE;
  float* fsum  = ws + o; o += (size_t)BB*128*SS;

  k_minmax <<<4, 256, 0, stream>>>(x, mm);
  k_knn1   <<<(BB*SS*32)/256, 256, 0, stream>>>(x, mm, idx1);
  k_pack   <<<(BB*PN)/256, 256, 0, stream>>>(x, pt);
  k_sel3   <<<(BB*SS)/256, 256, 0, stream>>>(x, idx1, sel3);
  k_knn2   <<<(BB*256)/4, 128, 0, stream>>>(pt, sel3, idx2);
  k_gather <<<(BB*NNE)/256, 256, 0, stream>>>(x, idx2, x1w);
  k_wpad   <<<2048/256, 256, 0, stream>>>(conv_w, pe_w, wpad);
  k_in16   <<<(BB*NNE)/256, 256, 0, stream>>>(x1w, in16);
  k_conv128<<<(BB*8*2048)/4, 128, 0, stream>>>(wpad, in16, feat);
  k_stats  <<<128, 256, 0, stream>>>(feat, stats);
  k_bn     <<<4096, 256, 0, stream>>>(feat, stats, bn_g, bn_b, bn2_g, bn2_b);
  k_att    <<<(BB*2048)/4, 128, 0, stream>>>(feat, att1_w, att1_b, att2_w, att2_b, logit);
  k_soft   <<<BB*SS, 128, 0, stream>>>(feat, logit, fsum);
  k_final  <<<(BB*8*256)/4, 128, 0, stream>>>(fsum, br1_w, br1_b, br2_w, br2_b, x1w, out);
  k_head   <<<(BB*5*SS)/256, 256, 0, stream>>>(x1w, out);
}